// VQ_47579647705783
// MI455X (gfx1250) — compile-verified
//
#include <hip/hip_runtime.h>

typedef __attribute__((ext_vector_type(2))) float v2f;
typedef __attribute__((ext_vector_type(8))) float v8f;

#define DIMS      100     // latent dims
#define NEMB      100     // embeddings
#define KPAD      112     // 7 tiles of 16 embeddings
#define ESTRIDE   102     // LDS row stride (even -> 8B-aligned float2, good bank spread)
#define NTILES    7

// Pass 1: squared norms of embeddings (padded entries poisoned so they never win argmin)
__global__ void vq_esq_kernel(const float* __restrict__ E, float* __restrict__ esq) {
    int k = threadIdx.x;
    if (k < KPAD) {
        float s = 1e30f;
        if (k < NEMB) {
            s = 0.0f;
            for (int d = 0; d < DIMS; ++d) { float e = E[k * DIMS + d]; s += e * e; }
        }
        esq[k] = s;
    }
}

// Pass 2: WMMA distance tiles + argmin + gather.
// Block = 256 threads = 8 waves; each wave owns 16 rows of X.
__global__ __launch_bounds__(256) void vq_main_kernel(
    const float* __restrict__ X, const float* __restrict__ E,
    const float* __restrict__ esq, float* __restrict__ out, int nTiles) {
    __shared__ float ldsE[KPAD * ESTRIDE];

    const int tid = threadIdx.x;

    // Stage embeddings into LDS (coalesced: E is 10000 contiguous floats)
    for (int i = tid; i < NEMB * DIMS; i += 256) {
        int r = i / DIMS;
        int c = i - r * DIMS;
        ldsE[r * ESTRIDE + c] = E[i];
    }
    // Zero the 12 padded embedding rows (rows 100..111 contiguous in LDS)
    for (int i = tid; i < (KPAD - NEMB) * ESTRIDE; i += 256) {
        ldsE[NEMB * ESTRIDE + i] = 0.0f;
    }
    __syncthreads();

    const int lane = tid & 31;
    const int wave = tid >> 5;
    const int rowTile = blockIdx.x * 8 + wave;
    if (rowTile >= nTiles) return;              // wave-uniform: EXEC stays all-ones for WMMA

    const int m    = lane & 15;                 // row-within-tile (A) / emb-within-tile (B)
    const int half = lane >> 4;
    const int kk   = half * 2;                  // K-subcolumn per 16x4 / 4x16 f32 layout
    const long rowBase = (long)rowTile * 16;

    // Per-lane e_sq for the 7 embedding tiles
    float eq[NTILES];
#pragma unroll
    for (int t = 0; t < NTILES; ++t) eq[t] = esq[t * 16 + m];

    v8f acc[NTILES] = {};

    // dot(X_row, E_k) accumulated via v_wmma_f32_16x16x4_f32, 25 K-steps
    const float* xrow = X + (rowBase + m) * DIMS;
    for (int k0 = 0; k0 < DIMS; k0 += 4) {
        v2f a = *(const v2f*)(xrow + k0 + kk);   // A[m, k0+2*half .. +1], 8B aligned
#pragma unroll
        for (int t = 0; t < NTILES; ++t) {
            v2f b = *(const v2f*)(&ldsE[(t * 16 + m) * ESTRIDE + k0 + kk]);
            acc[t] = __builtin_amdgcn_wmma_f32_16x16x4_f32(
                false, a, false, b, (short)0, acc[t], false, false);
        }
    }

    // Argmin of (e_sq - 2*dot): fold tiles in registers, then xor-reduce the 16-lane half.
    // C layout: VGPR r -> lanes 0-15 hold row r, lanes 16-31 hold row r+8.
    int myIdx = 0;
#pragma unroll
    for (int r = 0; r < 8; ++r) {
        float v = eq[0] - 2.0f * acc[0][r];
        int  bi = m;
#pragma unroll
        for (int t = 1; t < NTILES; ++t) {
            float vt = eq[t] - 2.0f * acc[t][r];
            int   it = t * 16 + m;
            if (vt < v) { v = vt; bi = it; }
        }
#pragma unroll
        for (int s = 1; s <= 8; s <<= 1) {
            float ov = __shfl_xor(v, s, 32);
            int   oi = __shfl_xor(bi, s, 32);
            if (ov < v || (ov == v && oi < bi)) { v = ov; bi = oi; }
        }
        if (m == r) myIdx = bi;   // lane r owns row r; lane 16+r owns row r+8
    }

    // Gather winners: 16 rows x 100 floats = 1600 contiguous output floats -> coalesced float2 stores
    float* orow = out + rowBase * DIMS;
    for (int j = lane; j < (16 * DIMS) / 2; j += 32) {   // 800 float2 elements
        int row = j / 50;
        int c2  = j - row * 50;
        int src = (row < 8) ? row : (row + 8);           // owning lane for this row
        int idx = __shfl(myIdx, src, 32);
        const float2 val = *(const float2*)&ldsE[idx * ESTRIDE + c2 * 2];
        *(float2*)&orow[j * 2] = val;
    }
}

extern "C" void kernel_launch(void* const* d_in, const int* in_sizes, int n_in,
                              void* d_out, int out_size, void* d_ws, size_t ws_size,
                              hipStream_t stream) {
    const float* X   = (const float*)d_in[0];   // [N, 100] fp32
    const float* E   = (const float*)d_in[1];   // [100, 100] fp32
    float*       out = (float*)d_out;           // [N, 100] fp32
    float*       esq = (float*)d_ws;            // 112 floats of scratch

    const int N      = in_sizes[0] / DIMS;      // 500000 (multiple of 16)
    const int nTiles = N / 16;                  // 31250

    vq_esq_kernel<<<1, 128, 0, stream>>>(E, esq);

    const int blocks = (nTiles + 7) / 8;        // 8 row-tiles (waves) per block
    vq_main_kernel<<<blocks, 256, 0, stream>>>(X, E, esq, out, nTiles);
}